// MathEmbedding_13580686590016
// MI455X (gfx1250) — compile-verified
//
#include <hip/hip_runtime.h>

// ---------------- problem constants ----------------
constexpr int kB      = 128;   // batch
constexpr int kS      = 256;   // sequence length
constexpr int kDepth  = 655;   // MathTypeKeysMax
constexpr int kU      = 128;   // UNITS
constexpr int kNG     = 512;   // 4*UNITS (gate width)
constexpr int kBuck   = 32;    // PROBLEM_BUCKETS
constexpr int kM      = kB * kS;          // 32768 rows of x / xz
constexpr int kCtx    = 35;               // 3 numeric + 32 one-hot

typedef __attribute__((ext_vector_type(2))) float v2f;
typedef __attribute__((ext_vector_type(8))) float v8f;

__device__ __forceinline__ float sigmf(float v) { return 1.0f / (1.0f + expf(-v)); }

// ---------------- kernel 1: context features (+ seq_len scalar) ----------------
__global__ void ctx_kernel(const int* __restrict__ last_rule,
                           const int* __restrict__ move_count,
                           const int* __restrict__ node_count,
                           const int* __restrict__ problem_type,
                           float* __restrict__ ctx,
                           int write_seq, float* __restrict__ seq_ptr) {
  int b = threadIdx.x;
  if (b < kB) {
    float* o = ctx + b * kCtx;
    o[0] = (float)last_rule[b];
    o[1] = (float)move_count[b];
    o[2] = (float)node_count[b];
    int p = problem_type[b];
#pragma unroll
    for (int j = 0; j < kBuck; ++j) o[3 + j] = (j == p) ? 1.0f : 0.0f;
  }
  if (threadIdx.x == 0 && write_seq) *seq_ptr = (float)kS;
}

// ---------------- kernel 2: one-hot "dense" == embedding gather ----------------
// x[m, u] = Wd[bwd[m]][u] + Wd[655+fwd[m]][u] + Wd[1310+lbwd[m]][u] + Wd[1965+lfwd[m]][u] + bd[u]
__global__ void __launch_bounds__(kU) gather_x_kernel(
    const int* __restrict__ bwd, const int* __restrict__ fwd,
    const int* __restrict__ lbwd, const int* __restrict__ lfwd,
    const float* __restrict__ Wd, const float* __restrict__ bd,
    float* __restrict__ x) {
  int m = blockIdx.x;        // 0 .. B*S-1
  int u = threadIdx.x;       // 0 .. 127
  int i0 = bwd[m];
  int i1 = kDepth + fwd[m];
  int i2 = 2 * kDepth + lbwd[m];
  int i3 = 3 * kDepth + lfwd[m];
  x[(size_t)m * kU + u] =
      Wd[(size_t)i0 * kU + u] + Wd[(size_t)i1 * kU + u] +
      Wd[(size_t)i2 * kU + u] + Wd[(size_t)i3 * kU + u] + bd[u];
}

// ---------------- kernel 3: transpose [128 x 512] -> [512 x 128] ----------------
__global__ void __launch_bounds__(kU) transpose_kn_kernel(const float* __restrict__ in,
                                                          float* __restrict__ out) {
  int n = blockIdx.x;   // 0..511
  int k = threadIdx.x;  // 0..127
  out[n * kU + k] = in[k * kNG + n];
}

// ---------------- kernel 4: xz = x @ Wx + b_lstm  (fp32 WMMA 16x16x4) ----------------
// Each wave computes one 16(M) x 32(N) tile = two 16x16 accumulators, K=128 in steps of 4.
// A frag (16x4 f32): lane L<16 -> M=L, holds K = k0+{0,1}; lane L>=16 -> M=L-16, K = k0+{2,3}.
// B frag (4x16 f32): lane L<16 -> N=L,  holds K = k0+{0,1}; lane L>=16 -> N=L-16, K = k0+{2,3}.
// (WxT is N-major so each lane's 2 K-values are one contiguous float2.)
__global__ void __launch_bounds__(256) gemm_xz_kernel(const float* __restrict__ x,
                                                      const float* __restrict__ WxT,
                                                      const float* __restrict__ bl,
                                                      float* __restrict__ xz) {
  int wave = (blockIdx.x * 256 + threadIdx.x) >> 5;  // 32768 wave-tiles
  int lane = threadIdx.x & 31;
  int tn = wave & 15;        // 16 N tiles of 32
  int tm = wave >> 4;        // 2048 M tiles of 16
  int m16 = tm * 16, n0 = tn * 32;
  int row  = lane & 15;
  int half = lane >> 4;
  int koff = half * 2;

  const float* arow = x   + (size_t)(m16 + row) * kU + koff;
  const float* bT0  = WxT + (size_t)(n0 + row) * kU + koff;
  const float* bT1  = WxT + (size_t)(n0 + 16 + row) * kU + koff;

  v8f acc0 = {}; v8f acc1 = {};
#pragma unroll 8
  for (int k0 = 0; k0 < kU; k0 += 4) {
    v2f a  = *(const v2f*)(arow + k0);
    v2f b0 = *(const v2f*)(bT0 + k0);
    v2f b1 = *(const v2f*)(bT1 + k0);
    acc0 = __builtin_amdgcn_wmma_f32_16x16x4_f32(false, a, false, b0, (short)0, acc0, false, false);
    acc1 = __builtin_amdgcn_wmma_f32_16x16x4_f32(false, a, false, b1, (short)0, acc1, false, false);
  }
  // D layout: vgpr r -> M = r + 8*half (lanes 0-15: M=r, lanes 16-31: M=r+8), N = lane&15
#pragma unroll
  for (int r = 0; r < 8; ++r) {
    int m = m16 + r + 8 * half;
    xz[(size_t)m * kNG + n0 + row]      = acc0[r] + bl[n0 + row];
    xz[(size_t)m * kNG + n0 + 16 + row] = acc1[r] + bl[n0 + 16 + row];
  }
}

// ---------------- kernel 5: LSTM scan ----------------
// 8 workgroups, each owns 16 batch rows (recurrence independent per row -> no inter-WG sync).
// 512 threads = 16 waves; wave w owns gate columns [32w, 32w+32).
// Per step: z[16x512] = xz[:,t,:] (WMMA C operand) + h[16x128] @ Wh[128x512], then gates.
// Critical-path fix: the C operand for step t+1 is prefetched into registers while the
// K-loop + gate phase of step t executes, hiding the L2/HBM round trip each serial step.
constexpr int kHP = kU + 4;    // h row stride pad: banks rotate by 4/row (kills conflicts)
constexpr int kZP = kNG + 4;   // z row stride pad: half-wave banks offset by 32

__global__ void __launch_bounds__(512) lstm_scan_kernel(const float* __restrict__ xz,
                                                        const float* __restrict__ WhT,
                                                        float* __restrict__ hs_out,
                                                        float* __restrict__ hlast_out) {
  __shared__ float h_lds[16][kHP];
  __shared__ float c_lds[16][kU];
  __shared__ float z_lds[16][kZP];

  int bb   = blockIdx.x;          // batch block: rows [16*bb, 16*bb+16)
  int tid  = threadIdx.x;
  int wave = tid >> 5;
  int lane = tid & 31;
  int row  = lane & 15;
  int half = lane >> 4;
  int koff = half * 2;
  int n0   = wave * 32;

  for (int e = tid; e < 16 * kU; e += 512) {
    h_lds[e >> 7][e & 127] = 0.0f;
    c_lds[e >> 7][e & 127] = 0.0f;
  }
  __syncthreads();

  const float* bT0 = WhT + (size_t)(n0 + row) * kU + koff;
  const float* bT1 = WhT + (size_t)(n0 + 16 + row) * kU + koff;

  // per-lane base pointers into xz for the C operand (C/D layout:
  // vgpr r -> batch-row r + 8*half, N = lane&15); advance by kNG per step
  const float* cbase[8];
#pragma unroll
  for (int r = 0; r < 8; ++r) {
    int b = bb * 16 + r + 8 * half;
    cbase[r] = xz + (size_t)b * kS * kNG;
  }

  // prefetch C for t = 0
  v8f nxt0, nxt1;
#pragma unroll
  for (int r = 0; r < 8; ++r) {
    nxt0[r] = cbase[r][n0 + row];
    nxt1[r] = cbase[r][n0 + 16 + row];
  }

  for (int t = 0; t < kS; ++t) {
    v8f acc0 = nxt0;
    v8f acc1 = nxt1;

    // issue next step's C loads now; wait lands at next iteration's copy
    size_t coff = (size_t)((t + 1 < kS) ? (t + 1) : t) * kNG;
#pragma unroll
    for (int r = 0; r < 8; ++r) {
      nxt0[r] = cbase[r][coff + n0 + row];
      nxt1[r] = cbase[r][coff + n0 + 16 + row];
    }

    // z += h @ Wh
#pragma unroll 8
    for (int k0 = 0; k0 < kU; k0 += 4) {
      v2f a  = *(const v2f*)(&h_lds[row][k0 + koff]);
      v2f b0 = *(const v2f*)(bT0 + k0);
      v2f b1 = *(const v2f*)(bT1 + k0);
      acc0 = __builtin_amdgcn_wmma_f32_16x16x4_f32(false, a, false, b0, (short)0, acc0, false, false);
      acc1 = __builtin_amdgcn_wmma_f32_16x16x4_f32(false, a, false, b1, (short)0, acc1, false, false);
    }
#pragma unroll
    for (int r = 0; r < 8; ++r) {
      int mr = r + 8 * half;
      z_lds[mr][n0 + row]      = acc0[r];
      z_lds[mr][n0 + 16 + row] = acc1[r];
    }
    __syncthreads();

    // gates: keras order i,f,g,o at columns u, u+128, u+256, u+384
    for (int e = tid; e < 16 * kU; e += 512) {
      int rr = e >> 7, u = e & 127;
      float gi = sigmf(z_lds[rr][u]);
      float gf = sigmf(z_lds[rr][u + kU]);
      float gg = tanhf(z_lds[rr][u + 2 * kU]);
      float go = sigmf(z_lds[rr][u + 3 * kU]);
      float cc = gf * c_lds[rr][u] + gi * gg;
      c_lds[rr][u] = cc;
      float hh = go * tanhf(cc);
      h_lds[rr][u] = hh;
      int b = bb * 16 + rr;
      hs_out[((size_t)b * kS + t) * kU + u] = hh;
      if (t == kS - 1) hlast_out[(size_t)b * kU + u] = hh;
    }
    __syncthreads();  // h/c ready before next step's matmul; z reads done before overwrite
  }
}

// ---------------- host launcher ----------------
extern "C" void kernel_launch(void* const* d_in, const int* in_sizes, int n_in,
                              void* d_out, int out_size, void* d_ws, size_t ws_size,
                              hipStream_t stream) {
  const int*   move_count   = (const int*)d_in[0];
  // d_in[1] = moves_remaining (unused by reference)
  const int*   last_rule    = (const int*)d_in[2];
  const int*   node_count   = (const int*)d_in[3];
  const int*   problem_type = (const int*)d_in[4];
  const int*   bwd          = (const int*)d_in[5];
  const int*   fwd          = (const int*)d_in[6];
  const int*   lbwd         = (const int*)d_in[7];
  const int*   lfwd         = (const int*)d_in[8];
  const float* Wd           = (const float*)d_in[9];
  const float* bd           = (const float*)d_in[10];
  const float* Wx           = (const float*)d_in[11];
  const float* Wh           = (const float*)d_in[12];
  const float* bl           = (const float*)d_in[13];

  // output tuple, flattened in return order
  float* out   = (float*)d_out;
  float* ctx   = out;                                   // [128, 35]
  float* hlast = out + (size_t)kB * kCtx;               // [128, 128]
  float* hs    = hlast + (size_t)kB * kU;               // [128, 256, 128]
  size_t seq_idx = (size_t)kB * kCtx + (size_t)kB * kU + (size_t)kB * kS * kU;

  // workspace: x [32768,128] | xz [32768,512] | WxT [512,128] | WhT [512,128]
  float* x   = (float*)d_ws;
  float* xz  = x + (size_t)kM * kU;
  float* WxT = xz + (size_t)kM * kNG;
  float* WhT = WxT + (size_t)kNG * kU;

  int write_seq = ((size_t)out_size > seq_idx) ? 1 : 0;

  ctx_kernel<<<1, 128, 0, stream>>>(last_rule, move_count, node_count, problem_type,
                                    ctx, write_seq, out + seq_idx);
  gather_x_kernel<<<kM, kU, 0, stream>>>(bwd, fwd, lbwd, lfwd, Wd, bd, x);
  transpose_kn_kernel<<<kNG, kU, 0, stream>>>(Wx, WxT);
  transpose_kn_kernel<<<kNG, kU, 0, stream>>>(Wh, WhT);
  gemm_xz_kernel<<<kM / (8 * 16), 256, 0, stream>>>(x, WxT, bl, xz);  // 256 blocks x 8 waves = 2048 wave-tiles
  lstm_scan_kernel<<<kB / 16, 512, 0, stream>>>(xz, WhT, hs, hlast);
}